// SetAbstraction_17214228922722
// MI455X (gfx1250) — compile-verified
//
#include <hip/hip_runtime.h>
#include <hip/hip_bf16.h>

typedef float v2f __attribute__((ext_vector_type(2)));
typedef float v8f __attribute__((ext_vector_type(8)));

#define BATCH 8
#define NPTS 4096
#define CIN 96
#define C2 192
#define NPOINT 1024
#define NSAMPLE 16
#define R2 0.01f
#define BN_EPS 1e-5f

// ---------------------------------------------------------------------------
// Kernel 1: furthest point sampling. One block per batch.
// ---------------------------------------------------------------------------
__global__ __launch_bounds__(256) void fps_kernel(const float* __restrict__ p,
                                                  int* __restrict__ samp) {
  const int b = blockIdx.x;
  const int tid = threadIdx.x;
  __shared__ float px[NPTS], py[NPTS], pz[NPTS];
  __shared__ float red_v[256];
  __shared__ int red_i[256];

  const float* pb = p + (size_t)b * NPTS * 3;
  for (int i = tid; i < NPTS; i += 256) {
    px[i] = pb[i * 3 + 0];
    py[i] = pb[i * 3 + 1];
    pz[i] = pb[i * 3 + 2];
  }
  float mind[16];
#pragma unroll
  for (int j = 0; j < 16; ++j) mind[j] = 1e10f;
  int last = 0;
  __syncthreads();

  for (int it = 0; it < NPOINT; ++it) {
    if (tid == 0) samp[b * NPOINT + it] = last;
    const float lx = px[last], ly = py[last], lz = pz[last];
    float bestv = -1.0f;
    int besti = NPTS;
#pragma unroll
    for (int j = 0; j < 16; ++j) {
      const int idx = j * 256 + tid;
      const float dx = px[idx] - lx, dy = py[idx] - ly, dz = pz[idx] - lz;
      const float d = dx * dx + dy * dy + dz * dz;
      const float m = fminf(mind[j], d);
      mind[j] = m;
      if (m > bestv || (m == bestv && idx < besti)) { bestv = m; besti = idx; }
    }
    red_v[tid] = bestv;
    red_i[tid] = besti;
    __syncthreads();
    for (int s = 128; s > 0; s >>= 1) {
      if (tid < s) {
        const float v2 = red_v[tid + s];
        const int i2 = red_i[tid + s];
        if (v2 > red_v[tid] || (v2 == red_v[tid] && i2 < red_i[tid])) {
          red_v[tid] = v2;
          red_i[tid] = i2;
        }
      }
      __syncthreads();
    }
    last = red_i[0];
    __syncthreads();
  }
}

// ---------------------------------------------------------------------------
// Kernel 2: ball query grouping. 256 centers per block; points staged in LDS.
// Writes new_p (into d_out head), nbr indices and dp relative coords (ws).
// ---------------------------------------------------------------------------
__global__ __launch_bounds__(256) void group_kernel(const float* __restrict__ p,
                                                    const int* __restrict__ samp,
                                                    int* __restrict__ nbr,
                                                    float* __restrict__ dp,
                                                    float* __restrict__ newp) {
  const int b = blockIdx.x / 4;
  const int gblk = blockIdx.x % 4;
  const int tid = threadIdx.x;
  __shared__ float px[NPTS], py[NPTS], pz[NPTS];
  const float* pb = p + (size_t)b * NPTS * 3;
  for (int i = tid; i < NPTS; i += 256) {
    px[i] = pb[i * 3 + 0];
    py[i] = pb[i * 3 + 1];
    pz[i] = pb[i * 3 + 2];
  }
  __syncthreads();

  const int g = gblk * 256 + tid;               // group id in [0,1024)
  const int sidx = samp[b * NPOINT + g];
  const float cx = px[sidx], cy = py[sidx], cz = pz[sidx];
  newp[(size_t)(b * NPOINT + g) * 3 + 0] = cx;
  newp[(size_t)(b * NPOINT + g) * 3 + 1] = cy;
  newp[(size_t)(b * NPOINT + g) * 3 + 2] = cz;

  int nb[NSAMPLE];
  int cnt = 0;
  for (int j = 0; j < NPTS && cnt < NSAMPLE; ++j) {
    const float dx = px[j] - cx, dy = py[j] - cy, dz = pz[j] - cz;
    if (dx * dx + dy * dy + dz * dz < R2) nb[cnt++] = j;
  }
  if (cnt == 0) nb[cnt++] = sidx;               // unreachable (center is in p)
  for (int k = cnt; k < NSAMPLE; ++k) nb[k] = nb[0];

  const size_t base = (size_t)(b * NPOINT + g) * NSAMPLE;
  const size_t dbase = ((size_t)b * 3 * NPOINT + g) * NSAMPLE;
#pragma unroll
  for (int k = 0; k < NSAMPLE; ++k) {
    const int j = nb[k];
    nbr[base + k] = j;
    dp[dbase + k] = px[j] - cx;
    dp[dbase + (size_t)NPOINT * NSAMPLE + k] = py[j] - cy;
    dp[dbase + (size_t)2 * NPOINT * NSAMPLE + k] = pz[j] - cz;
  }
}

// ---------------------------------------------------------------------------
// Kernel 3: transfer matmul (Wt @ relu(xj)) + BN + ReLU + PosPool + pooling.
// 12 waves/block = 12 M-tiles of 16 output channels. 4 groups per block.
// WMMA: f32 16x16x4, K-loop of 48 with weights resident in 96 VGPRs.
// ---------------------------------------------------------------------------
#define GP 4
#define XJPAD 72
__global__ __launch_bounds__(384) void transfer_kernel(
    const float* __restrict__ x, const int* __restrict__ samp,
    const int* __restrict__ nbr, const float* __restrict__ dp,
    const float* __restrict__ Wt, const float* __restrict__ gt,
    const float* __restrict__ bt, const float* __restrict__ mt,
    const float* __restrict__ vt, float* __restrict__ pooled) {
  const int b = blockIdx.x / (NPOINT / GP);
  const int g0 = (blockIdx.x % (NPOINT / GP)) * GP;
  const int tid = threadIdx.x;

  __shared__ float xj[C2][XJPAD];               // [channel][g*16+s], padded
  __shared__ float dps[3][GP][NSAMPLE];

  for (int i = tid; i < 3 * GP * NSAMPLE; i += 384) {
    const int coord = i / (GP * NSAMPLE);
    const int rem = i % (GP * NSAMPLE);
    dps[coord][rem / NSAMPLE][rem % NSAMPLE] =
        dp[(((size_t)b * 3 + coord) * NPOINT + g0 + rem / NSAMPLE) * NSAMPLE +
           rem % NSAMPLE];
  }
  for (int i = tid; i < C2 * GP * NSAMPLE; i += 384) {
    const int c = i / (GP * NSAMPLE);
    const int rem = i % (GP * NSAMPLE);
    const int gg = g0 + rem / NSAMPLE;
    const int s = rem % NSAMPLE;
    int n;
    int cc;
    if (c < CIN) {
      n = samp[b * NPOINT + gg];
      cc = c;
    } else {
      n = nbr[(size_t)(b * NPOINT + gg) * NSAMPLE + s];
      cc = c - CIN;
    }
    const float v = x[((size_t)b * CIN + cc) * NPTS + n];
    xj[c][rem] = fmaxf(v, 0.0f);                // relu before transfer conv
  }
  __syncthreads();

  const int wave = tid >> 5;                    // 0..11 -> M tile
  const int lane = tid & 31;
  const int half = lane >> 4;
  const int ln = lane & 15;
  const int m0 = wave * 16;

  // A fragments: rows m0..m0+15 of Wt, K = 4k + 2*half + {0,1}
  v2f Areg[48];
  const int arow = m0 + ln;
#pragma unroll
  for (int k = 0; k < 48; ++k) {
    const int c = 4 * k + 2 * half;
    Areg[k].x = Wt[arow * C2 + c];
    Areg[k].y = Wt[arow * C2 + c + 1];
  }
  // BN folding + PosPool constants for the 8 output rows this lane owns
  float bscale[8], bbias[8], escale[8];
  int issin[8], coordv[8];
#pragma unroll
  for (int v = 0; v < 8; ++v) {
    const int o = m0 + v + 8 * half;
    const float sc = gt[o] * rsqrtf(vt[o] + BN_EPS);
    bscale[v] = sc;
    bbias[v] = bt[o] - mt[o] * sc;
    const int j = o & 63;
    const int jj = j & 31;
    escale[v] = 50.0f * powf(500.0f, -(float)jj * (1.0f / 32.0f));
    issin[v] = (j < 32);
    coordv[v] = o >> 6;
  }

  for (int g = 0; g < GP; ++g) {
    v8f acc = {};
#pragma unroll
    for (int k = 0; k < 48; ++k) {
      const int c = 4 * k + 2 * half;
      v2f bfrag;
      bfrag.x = xj[c][g * NSAMPLE + ln];
      bfrag.y = xj[c + 1][g * NSAMPLE + ln];
      acc = __builtin_amdgcn_wmma_f32_16x16x4_f32(false, Areg[k], false, bfrag,
                                                  (short)0, acc, false, false);
    }
#pragma unroll
    for (int v = 0; v < 8; ++v) {
      const int o = m0 + v + 8 * half;
      const float y = fmaxf(acc[v] * bscale[v] + bbias[v], 0.0f);
      const float theta = dps[coordv[v]][g][ln] * escale[v];
      const float e = issin[v] ? __sinf(theta) : __cosf(theta);
      const float a = (y + 1.0f) * e;            // y*emb + emb
      float mx = a, sm = a;                      // pool over 16 samples (lanes)
#pragma unroll
      for (int off = 1; off < 16; off <<= 1) {
        mx = fmaxf(mx, __shfl_xor(mx, off, 32));
        sm += __shfl_xor(sm, off, 32);
      }
      if (ln == 0)
        pooled[((size_t)b * C2 + o) * NPOINT + g0 + g] = mx + sm * (1.0f / 16.0f);
    }
  }
}

// ---------------------------------------------------------------------------
// Kernel 4: classifier matmul (Wc @ relu(pooled)) + BN + ReLU -> d_out tail.
// 64 positions per block, 12 waves = 12 M tiles, 4 N tiles.
// ---------------------------------------------------------------------------
__global__ __launch_bounds__(384) void classifier_kernel(
    const float* __restrict__ pooled, const float* __restrict__ Wc,
    const float* __restrict__ gc, const float* __restrict__ bc,
    const float* __restrict__ mc, const float* __restrict__ vc,
    float* __restrict__ out) {
  const int b = blockIdx.x / 16;
  const int q0 = (blockIdx.x % 16) * 64;
  const int tid = threadIdx.x;

  __shared__ float xt[C2][XJPAD];
  for (int i = tid; i < C2 * 64; i += 384) {
    const int c = i / 64, q = i % 64;
    xt[c][q] = fmaxf(pooled[((size_t)b * C2 + c) * NPOINT + q0 + q], 0.0f);
  }
  __syncthreads();

  const int wave = tid >> 5;
  const int lane = tid & 31;
  const int half = lane >> 4;
  const int ln = lane & 15;
  const int m0 = wave * 16;

  v2f Areg[48];
  const int arow = m0 + ln;
#pragma unroll
  for (int k = 0; k < 48; ++k) {
    const int c = 4 * k + 2 * half;
    Areg[k].x = Wc[arow * C2 + c];
    Areg[k].y = Wc[arow * C2 + c + 1];
  }
  float bscale[8], bbias[8];
#pragma unroll
  for (int v = 0; v < 8; ++v) {
    const int o = m0 + v + 8 * half;
    const float sc = gc[o] * rsqrtf(vc[o] + BN_EPS);
    bscale[v] = sc;
    bbias[v] = bc[o] - mc[o] * sc;
  }

  for (int t = 0; t < 4; ++t) {
    v8f acc = {};
#pragma unroll
    for (int k = 0; k < 48; ++k) {
      const int c = 4 * k + 2 * half;
      v2f bfrag;
      bfrag.x = xt[c][t * 16 + ln];
      bfrag.y = xt[c + 1][t * 16 + ln];
      acc = __builtin_amdgcn_wmma_f32_16x16x4_f32(false, Areg[k], false, bfrag,
                                                  (short)0, acc, false, false);
    }
#pragma unroll
    for (int v = 0; v < 8; ++v) {
      const int o = m0 + v + 8 * half;
      const float y = fmaxf(acc[v] * bscale[v] + bbias[v], 0.0f);
      out[((size_t)b * C2 + o) * NPOINT + q0 + t * 16 + ln] = y;
    }
  }
}

// ---------------------------------------------------------------------------
extern "C" void kernel_launch(void* const* d_in, const int* in_sizes, int n_in,
                              void* d_out, int out_size, void* d_ws,
                              size_t ws_size, hipStream_t stream) {
  (void)in_sizes; (void)n_in; (void)out_size; (void)ws_size;
  const float* p  = (const float*)d_in[0];
  const float* x  = (const float*)d_in[1];
  const float* Wt = (const float*)d_in[2];
  const float* gt = (const float*)d_in[3];
  const float* bt = (const float*)d_in[4];
  const float* mt = (const float*)d_in[5];
  const float* vt = (const float*)d_in[6];
  const float* Wc = (const float*)d_in[7];
  const float* gc = (const float*)d_in[8];
  const float* bc = (const float*)d_in[9];
  const float* mc = (const float*)d_in[10];
  const float* vc = (const float*)d_in[11];

  char* ws = (char*)d_ws;
  int*   samp   = (int*)ws;                                     // 8*1024 ints
  int*   nbr    = (int*)(ws + 32768);                           // 8*1024*16 ints
  float* dp     = (float*)(ws + 32768 + 524288);                // 8*3*1024*16 f32
  float* pooled = (float*)(ws + 32768 + 524288 + 1572864);      // 8*192*1024 f32

  float* newp_out = (float*)d_out;                              // [B,NP,3]
  float* out_tail = (float*)d_out + (size_t)BATCH * NPOINT * 3; // [B,C2,NP]

  fps_kernel<<<BATCH, 256, 0, stream>>>(p, samp);
  group_kernel<<<BATCH * 4, 256, 0, stream>>>(p, samp, nbr, dp, newp_out);
  transfer_kernel<<<BATCH * (NPOINT / GP), 384, 0, stream>>>(
      x, samp, nbr, dp, Wt, gt, bt, mt, vt, pooled);
  classifier_kernel<<<BATCH * 16, 384, 0, stream>>>(pooled, Wc, gc, bc, mc, vc,
                                                    out_tail);
}